// Sonata3DEncoder_171798692048
// MI455X (gfx1250) — compile-verified
//
#include <hip/hip_runtime.h>

typedef float v2f __attribute__((ext_vector_type(2)));
typedef float v8f __attribute__((ext_vector_type(8)));

#define NPTS   120000
#define NSEG   4096      // 8 batches * 512 segments
#define KDIM   1232      // 48 + 96 + 192 + 384 + 512
#define HDIM   256
#define LDSK   1236      // padded stride: 1236 % 64 = 20 -> rows hit distinct banks
#define MROWS  32        // M rows per block (2 WMMA M-tiles, B reused across both)
#define LN_EPS 1e-5f

// ---------------------------------------------------------------------------
// Kernel 1: zero the small integer workspace (cnt + first NSEG of offs; fill not covered).
// ---------------------------------------------------------------------------
__global__ void zero_int_kernel(int* __restrict__ ws, int n) {
    int i = blockIdx.x * blockDim.x + threadIdx.x;
    if (i < n) ws[i] = 0;
}

// ---------------------------------------------------------------------------
// Kernel 2: histogram of points per segment (120K int atomics, L2-resident).
// ---------------------------------------------------------------------------
__global__ __launch_bounds__(256) void hist_kernel(
    const int* __restrict__ batch, const int* __restrict__ p2s,
    int* __restrict__ cnt)
{
    int p = blockIdx.x * blockDim.x + threadIdx.x;
    if (p >= NPTS) return;
    int s = batch[p] * 512 + p2s[p];
    atomicAdd(&cnt[s], 1);
}

// ---------------------------------------------------------------------------
// Kernel 3: exclusive prefix scan over the 4096 segment counts (one block).
// ---------------------------------------------------------------------------
__global__ __launch_bounds__(1024) void scan_kernel(
    const int* __restrict__ cnt, int* __restrict__ offs)
{
    __shared__ int part[1024];
    const int t = threadIdx.x;
    int v0 = cnt[t * 4 + 0], v1 = cnt[t * 4 + 1];
    int v2 = cnt[t * 4 + 2], v3 = cnt[t * 4 + 3];
    int e1 = v0, e2 = v0 + v1, e3 = v0 + v1 + v2;
    int tot = e3 + v3;
    part[t] = tot;
    __syncthreads();
    for (int off = 1; off < 1024; off <<= 1) {
        int x = (t >= off) ? part[t - off] : 0;
        __syncthreads();
        part[t] += x;
        __syncthreads();
    }
    int ex = (t == 0) ? 0 : part[t - 1];
    offs[t * 4 + 0] = ex;
    offs[t * 4 + 1] = ex + e1;
    offs[t * 4 + 2] = ex + e2;
    offs[t * 4 + 3] = ex + e3;
    if (t == 1023) offs[4096] = ex + tot;
}

// ---------------------------------------------------------------------------
// Kernel 4: scatter point ids into per-segment lists.
// ---------------------------------------------------------------------------
__global__ __launch_bounds__(256) void fill_kernel(
    const int* __restrict__ batch, const int* __restrict__ p2s,
    const int* __restrict__ offs, int* __restrict__ fill,
    int* __restrict__ plist)
{
    int p = blockIdx.x * blockDim.x + threadIdx.x;
    if (p >= NPTS) return;
    int s = batch[p] * 512 + p2s[p];
    int pos = atomicAdd(&fill[s], 1);
    plist[offs[s] + pos] = p;
}

// ---------------------------------------------------------------------------
// Kernel 5: per-segment mean of the hierarchically gathered 1232-dim feature.
// One block (64 threads) per segment; 20 register accumulators per thread
// statically mapped onto the five level slices. No atomics at all.
// ---------------------------------------------------------------------------
__global__ __launch_bounds__(64) void segmean_kernel(
    const float* __restrict__ f0, const float* __restrict__ f1,
    const float* __restrict__ f2, const float* __restrict__ f3,
    const float* __restrict__ f4,
    const int* __restrict__ inv0, const int* __restrict__ inv1,
    const int* __restrict__ inv2, const int* __restrict__ inv3,
    const int* __restrict__ offs, const int* __restrict__ plist,
    float* __restrict__ mean)
{
    const int s   = blockIdx.x;
    const int t   = threadIdx.x;           // 0..63
    const int beg = offs[s];
    const int end = offs[s + 1];

    float a0 = 0.0f;
    float a1[2] = {};            // dims t, t+64   of 96
    float a2[3] = {};            // dims t+64j     of 192
    float a3[6] = {};            // dims t+64j     of 384
    float a4[8] = {};            // dims t+64j     of 512

    for (int i = beg; i < end; ++i) {
        int p  = plist[i];                 // uniform across the block
        int i1 = inv0[p];
        int i2 = inv1[i1];
        int i3 = inv2[i2];
        int i4 = inv3[i3];
        const float* r0 = f0 + (size_t)p  * 48;
        const float* r1 = f1 + (size_t)i1 * 96;
        const float* r2 = f2 + (size_t)i2 * 192;
        const float* r3 = f3 + (size_t)i3 * 384;
        const float* r4 = f4 + (size_t)i4 * 512;

        if (t < 48) a0 += r0[t];
        a1[0] += r1[t];
        if (t < 32) a1[1] += r1[t + 64];
#pragma unroll
        for (int j = 0; j < 3; ++j) a2[j] += r2[t + 64 * j];
#pragma unroll
        for (int j = 0; j < 6; ++j) a3[j] += r3[t + 64 * j];
#pragma unroll
        for (int j = 0; j < 8; ++j) a4[j] += r4[t + 64 * j];
    }

    const float scale = 1.0f / fmaxf((float)(end - beg), 1.0f);
    float* dst = mean + (size_t)s * KDIM;
    if (t < 48) dst[t] = a0 * scale;
    dst[48 + t] = a1[0] * scale;
    if (t < 32) dst[48 + 64 + t] = a1[1] * scale;
#pragma unroll
    for (int j = 0; j < 3; ++j) dst[144 + 64 * j + t] = a2[j] * scale;
#pragma unroll
    for (int j = 0; j < 6; ++j) dst[336 + 64 * j + t] = a3[j] * scale;
#pragma unroll
    for (int j = 0; j < 8; ++j) dst[720 + 64 * j + t] = a4[j] * scale;
}

// ---------------------------------------------------------------------------
// Kernel 6: mean @ W + b -> LayerNorm, using V_WMMA_F32_16X16X4_F32.
// Block = 128 threads (4 waves); 32 segment rows per block (2 M-tiles),
// each wave owns 4 N-tiles of 16 -> full N=256. Each B fragment feeds two
// WMMAs (both M-tiles), halving the L2 traffic on W. A staged in padded LDS.
// ---------------------------------------------------------------------------
__global__ __launch_bounds__(128) void gemm_ln_kernel(
    const float* __restrict__ mean, const float* __restrict__ W,
    const float* __restrict__ bias, const float* __restrict__ gamma,
    const float* __restrict__ beta, float* __restrict__ out)
{
    __shared__ alignas(16) float smean[MROWS * LDSK];   // 158208 B
    __shared__ float sx[MROWS * HDIM];                  //  32768 B
    __shared__ float smu[MROWS];
    __shared__ float srs[MROWS];

    const int tid = threadIdx.x;
    const int m0  = blockIdx.x * MROWS;

    // Stage the 32 x 1232 mean tile into LDS with 128-bit copies.
    for (int r = 0; r < MROWS; ++r) {
        const float4* src = (const float4*)(mean + (size_t)(m0 + r) * KDIM);
        float4* dst = (float4*)(smean + r * LDSK);
        for (int k4 = tid; k4 < KDIM / 4; k4 += 128) dst[k4] = src[k4];
    }
    __syncthreads();

    const int wave  = tid >> 5;
    const int lane  = tid & 31;
    const int arow  = lane & 15;          // M row for A operand
    const int khalf = (lane >> 4) << 1;   // lanes 16-31 carry K+2, K+3
    const int ncol  = wave * 64 + (lane & 15);

    v8f acc[2][4] = {};

    for (int k0 = 0; k0 < KDIM; k0 += 4) {
        const int kk = k0 + khalf;
        // A fragments for both M-tiles (LDS, bank-conflict-free stride)
        v2f a0, a1;
        a0.x = smean[arow * LDSK + kk];
        a0.y = smean[arow * LDSK + kk + 1];
        a1.x = smean[(16 + arow) * LDSK + kk];
        a1.y = smean[(16 + arow) * LDSK + kk + 1];
#pragma unroll
        for (int j = 0; j < 4; ++j) {
            const int n = ncol + j * 16;
            v2f bmat;
            bmat.x = W[(size_t)kk * HDIM + n];
            bmat.y = W[(size_t)(kk + 1) * HDIM + n];
            acc[0][j] = __builtin_amdgcn_wmma_f32_16x16x4_f32(
                false, a0, false, bmat, (short)0, acc[0][j], false, false);
            acc[1][j] = __builtin_amdgcn_wmma_f32_16x16x4_f32(
                false, a1, false, bmat, (short)0, acc[1][j], false, false);
        }
    }

    // Accumulator layout: VGPR i holds (M = mt*16 + i + 8*(lane>=16), N = ncol_j).
    const int rbase = (lane >> 4) << 3;
#pragma unroll
    for (int mt = 0; mt < 2; ++mt) {
#pragma unroll
        for (int j = 0; j < 4; ++j) {
            const int n   = ncol + j * 16;
            const float bv = bias[n];
#pragma unroll
            for (int i = 0; i < 8; ++i) {
                sx[(mt * 16 + rbase + i) * HDIM + n] = acc[mt][j][i] + bv;
            }
        }
    }
    __syncthreads();

    // LayerNorm stats: 128 threads = 32 rows x 4 partials of 64 elements.
    {
        const int r  = tid >> 2;
        const int c0 = (tid & 3) * 64;
        float s1 = 0.0f, s2 = 0.0f;
        const float* rowp = sx + r * HDIM + c0;
        for (int c = 0; c < 64; ++c) { float v = rowp[c]; s1 += v; s2 += v * v; }
        s1 += __shfl_xor(s1, 1, 32);  s2 += __shfl_xor(s2, 1, 32);
        s1 += __shfl_xor(s1, 2, 32);  s2 += __shfl_xor(s2, 2, 32);
        if ((tid & 3) == 0) {
            float mu  = s1 * (1.0f / 256.0f);
            float var = s2 * (1.0f / 256.0f) - mu * mu;
            smu[r] = mu;
            srs[r] = rsqrtf(var + LN_EPS);
        }
    }
    __syncthreads();

    for (int idx = tid; idx < MROWS * HDIM; idx += 128) {
        int r = idx >> 8;
        int n = idx & 255;
        float v = (sx[idx] - smu[r]) * srs[r] * gamma[n] + beta[n];
        out[(size_t)(m0 + r) * HDIM + n] = v;
    }
}

// ---------------------------------------------------------------------------
extern "C" void kernel_launch(void* const* d_in, const int* in_sizes, int n_in,
                              void* d_out, int out_size, void* d_ws, size_t ws_size,
                              hipStream_t stream) {
    const float* f0    = (const float*)d_in[0];
    const float* f1    = (const float*)d_in[1];
    const float* f2    = (const float*)d_in[2];
    const float* f3    = (const float*)d_in[3];
    const float* f4    = (const float*)d_in[4];
    const float* W     = (const float*)d_in[5];
    const float* bias  = (const float*)d_in[6];
    const float* gamma = (const float*)d_in[7];
    const float* beta  = (const float*)d_in[8];
    const int*   inv0  = (const int*)d_in[9];
    const int*   inv1  = (const int*)d_in[10];
    const int*   inv2  = (const int*)d_in[11];
    const int*   inv3  = (const int*)d_in[12];
    const int*   batch = (const int*)d_in[13];
    const int*   p2s   = (const int*)d_in[14];

    // Workspace layout.
    float* mean  = (float*)d_ws;                     // NSEG*KDIM floats
    int*   cnt   = (int*)(mean + (size_t)NSEG * KDIM);
    int*   offs  = cnt + NSEG;                       // NSEG + 1
    int*   fill  = offs + NSEG + 1;                  // NSEG
    int*   plist = fill + NSEG;                      // NPTS
    float* out   = (float*)d_out;

    zero_int_kernel<<<(2 * NSEG + 255) / 256, 256, 0, stream>>>(cnt, 2 * NSEG + 1);
    hist_kernel<<<(NPTS + 255) / 256, 256, 0, stream>>>(batch, p2s, cnt);
    scan_kernel<<<1, 1024, 0, stream>>>(cnt, offs);
    fill_kernel<<<(NPTS + 255) / 256, 256, 0, stream>>>(batch, p2s, offs, fill, plist);
    segmean_kernel<<<NSEG, 64, 0, stream>>>(
        f0, f1, f2, f3, f4, inv0, inv1, inv2, inv3, offs, plist, mean);
    gemm_ln_kernel<<<NSEG / MROWS, 128, 0, stream>>>(
        mean, W, bias, gamma, beta, out);
}